// MPMDiffSim_4166118277760
// MI455X (gfx1250) — compile-verified
//
#include <hip/hip_runtime.h>
#include <hip/hip_bf16.h>
#include <stdint.h>

// ---- MPM constants (match reference) ----
#define NG   128
#define NG3  (NG*NG*NG)

constexpr float DT_C      = 0.0002f;
constexpr float INV_DX_C  = 128.0f;
constexpr float DX_C      = 1.0f / 128.0f;
constexpr float P_VOL_C   = (0.5f/128.0f)*(0.5f/128.0f)*(0.5f/128.0f);
constexpr float P_MASS_C  = P_VOL_C * 1000.0f;
constexpr float STRESS_K  = -DT_C * P_VOL_C * 4.0f * INV_DX_C * INV_DX_C;
constexpr float GRAV_C    = 9.8f;
constexpr int   BOUND_C   = 3;

// ---------------- kernel 1: zero the grid (32 MB, lives in L2) ----------------
// NG3 is an exact multiple of 256 -> no bounds branch, EXEC all-ones.
__global__ void __launch_bounds__(256) mpm_zero(float4* __restrict__ g) {
    int i = blockIdx.x * blockDim.x + threadIdx.x;
    g[i] = make_float4(0.f, 0.f, 0.f, 0.f);
}

// ---------------- kernel 2: P2G scatter (atomic f32 adds into L2) --------------
__global__ void __launch_bounds__(256) mpm_p2g(const float* __restrict__ x,
                                               const float* __restrict__ v,
                                               const float* __restrict__ C,
                                               const float* __restrict__ stress,
                                               float* __restrict__ grid, int n) {
    int p = blockIdx.x * blockDim.x + threadIdx.x;
    if (p >= n) return;

    // prefetch the next tile of the streaming particle arrays (global_prefetch_b8)
    __builtin_prefetch(stress + (size_t)(p + 2048) * 9, 0, 0);
    __builtin_prefetch(C      + (size_t)(p + 2048) * 9, 0, 0);

    float xp[3], vp[3];
    #pragma unroll
    for (int d = 0; d < 3; ++d) { xp[d] = x[(size_t)p*3 + d]; vp[d] = v[(size_t)p*3 + d]; }

    float aff[3][3];
    #pragma unroll
    for (int i = 0; i < 3; ++i)
        #pragma unroll
        for (int j = 0; j < 3; ++j) {
            float s = stress[(size_t)p*9 + i*3 + j];
            float c = C[(size_t)p*9 + i*3 + j];
            aff[i][j] = STRESS_K * s + P_MASS_C * c;
        }

    int   base[3];
    float fx[3], w[3][3];
    #pragma unroll
    for (int d = 0; d < 3; ++d) {
        float gx = xp[d] * INV_DX_C;
        float bf = floorf(gx - 0.5f);
        base[d]  = (int)bf;
        fx[d]    = gx - bf;
        float a = 1.5f - fx[d];
        float b = fx[d] - 1.0f;
        float c = fx[d] - 0.5f;
        w[0][d] = 0.5f * a * a;
        w[1][d] = 0.75f - b * b;
        w[2][d] = 0.5f * c * c;
    }

    const float mvb[3] = { P_MASS_C*vp[0], P_MASS_C*vp[1], P_MASS_C*vp[2] };

    #pragma unroll
    for (int i = 0; i < 3; ++i) {
        int ni = min(max(base[0] + i, 0), NG - 1);
        float dpx = ((float)i - fx[0]) * DX_C;
        float mvi[3];
        #pragma unroll
        for (int d = 0; d < 3; ++d) mvi[d] = mvb[d] + aff[d][0] * dpx;
        #pragma unroll
        for (int j = 0; j < 3; ++j) {
            int nj = min(max(base[1] + j, 0), NG - 1);
            float dpy = ((float)j - fx[1]) * DX_C;
            float wij = w[i][0] * w[j][1];
            float mvij[3];
            #pragma unroll
            for (int d = 0; d < 3; ++d) mvij[d] = mvi[d] + aff[d][1] * dpy;
            int rowflat = (ni * NG + nj) * NG;
            #pragma unroll
            for (int k = 0; k < 3; ++k) {
                int nk = min(max(base[2] + k, 0), NG - 1);
                float dpz = ((float)k - fx[2]) * DX_C;
                float wt  = wij * w[k][2];
                float* g  = grid + (size_t)(rowflat + nk) * 4;
                atomicAdd(g + 0, wt * (mvij[0] + aff[0][2] * dpz));
                atomicAdd(g + 1, wt * (mvij[1] + aff[1][2] * dpz));
                atomicAdd(g + 2, wt * (mvij[2] + aff[2][2] * dpz));
                atomicAdd(g + 3, wt * P_MASS_C);
            }
        }
    }
}

// ---------------- kernel 3: grid normalize + gravity + boundary ----------------
// Streaming pass staged through LDS with the CDNA5 async global->LDS engine:
// global_load_async_to_lds_b128 (tracked on ASYNCcnt) + s_wait_asynccnt + ds read.
__global__ void __launch_bounds__(256) mpm_grid_update(float4* __restrict__ grid) {
    __shared__ float4 tile[256];
    int idx = blockIdx.x * 256 + threadIdx.x;

    // LDS byte address = low 32 bits of the flat shared-aperture pointer.
    uint32_t lds_addr = (uint32_t)(uintptr_t)&tile[threadIdx.x];
    uint64_t gaddr    = (uint64_t)(uintptr_t)(grid + idx);

    asm volatile(
        "global_load_async_to_lds_b128 %0, %1, off\n\t"
        "s_wait_asynccnt 0x0"
        :
        : "v"(lds_addr), "v"(gaddr)
        : "memory");

    float4 g = tile[threadIdx.x];   // ds_load_b128

    float m = g.w;
    float gv[3] = {0.f, 0.f, 0.f};
    if (m > 0.f) {
        float inv = 1.0f / fmaxf(m, 1e-10f);
        gv[0] = g.x * inv;
        gv[1] = g.y * inv - DT_C * GRAV_C;
        gv[2] = g.z * inv;
    }
    int k = idx & (NG - 1);
    int j = (idx >> 7) & (NG - 1);
    int i = idx >> 14;
    if (i <  BOUND_C      && gv[0] < 0.f) gv[0] = 0.f;
    if (i >= NG - BOUND_C && gv[0] > 0.f) gv[0] = 0.f;
    if (j <  BOUND_C      && gv[1] < 0.f) gv[1] = 0.f;
    if (j >= NG - BOUND_C && gv[1] > 0.f) gv[1] = 0.f;
    if (k <  BOUND_C      && gv[2] < 0.f) gv[2] = 0.f;
    if (k >= NG - BOUND_C && gv[2] > 0.f) gv[2] = 0.f;
    g.x = gv[0]; g.y = gv[1]; g.z = gv[2];
    grid[idx] = g;
}

// ---------------- kernel 4: G2P gather + particle update -----------------------
__global__ void __launch_bounds__(256) mpm_g2p(const float* __restrict__ x,
                                               const float* __restrict__ Fin,
                                               const float4* __restrict__ grid,
                                               float* __restrict__ out, int n) {
    int p = blockIdx.x * blockDim.x + threadIdx.x;
    if (p >= n) return;

    __builtin_prefetch(Fin + (size_t)(p + 2048) * 9, 0, 0);

    float xp[3];
    #pragma unroll
    for (int d = 0; d < 3; ++d) xp[d] = x[(size_t)p*3 + d];

    int   base[3];
    float fx[3], w[3][3];
    #pragma unroll
    for (int d = 0; d < 3; ++d) {
        float gx = xp[d] * INV_DX_C;
        float bf = floorf(gx - 0.5f);
        base[d]  = (int)bf;
        fx[d]    = gx - bf;
        float a = 1.5f - fx[d];
        float b = fx[d] - 1.0f;
        float c = fx[d] - 0.5f;
        w[0][d] = 0.5f * a * a;
        w[1][d] = 0.75f - b * b;
        w[2][d] = 0.5f * c * c;
    }

    float vnext[3] = {0.f, 0.f, 0.f};
    float B[3][3]  = {{0.f,0.f,0.f},{0.f,0.f,0.f},{0.f,0.f,0.f}};

    #pragma unroll
    for (int i = 0; i < 3; ++i) {
        int ni = min(max(base[0] + i, 0), NG - 1);
        float dpx = ((float)i - fx[0]) * DX_C;
        #pragma unroll
        for (int j = 0; j < 3; ++j) {
            int nj = min(max(base[1] + j, 0), NG - 1);
            float dpy = ((float)j - fx[1]) * DX_C;
            float wij = w[i][0] * w[j][1];
            int rowflat = (ni * NG + nj) * NG;
            #pragma unroll
            for (int k = 0; k < 3; ++k) {
                int nk = min(max(base[2] + k, 0), NG - 1);
                float dpz = ((float)k - fx[2]) * DX_C;
                float wt  = wij * w[k][2];
                float4 g  = grid[rowflat + nk];      // global_load_b128, L2-resident
                float gv0 = g.x, gv1 = g.y, gv2 = g.z;
                vnext[0] += wt * gv0;
                vnext[1] += wt * gv1;
                vnext[2] += wt * gv2;
                B[0][0] += wt * gv0 * dpx; B[0][1] += wt * gv0 * dpy; B[0][2] += wt * gv0 * dpz;
                B[1][0] += wt * gv1 * dpx; B[1][1] += wt * gv1 * dpy; B[1][2] += wt * gv1 * dpz;
                B[2][0] += wt * gv2 * dpx; B[2][1] += wt * gv2 * dpy; B[2][2] += wt * gv2 * dpz;
            }
        }
    }

    const float k4 = 4.0f * INV_DX_C * INV_DX_C;
    float Cn[3][3], A[3][3];
    #pragma unroll
    for (int i = 0; i < 3; ++i)
        #pragma unroll
        for (int j = 0; j < 3; ++j) {
            Cn[i][j] = k4 * B[i][j];
            A[i][j]  = DT_C * Cn[i][j] + (i == j ? 1.0f : 0.0f);
        }

    float Fp[3][3];
    #pragma unroll
    for (int i = 0; i < 3; ++i)
        #pragma unroll
        for (int j = 0; j < 3; ++j)
            Fp[i][j] = Fin[(size_t)p*9 + i*3 + j];

    float Fn[3][3];
    #pragma unroll
    for (int i = 0; i < 3; ++i)
        #pragma unroll
        for (int kk = 0; kk < 3; ++kk)
            Fn[i][kk] = A[i][0]*Fp[0][kk] + A[i][1]*Fp[1][kk] + A[i][2]*Fp[2][kk];

    // output layout: [x (3n) | v (3n) | C (9n) | F (9n)]
    size_t N = (size_t)n;
    float* out_x = out;
    float* out_v = out + 3*N;
    float* out_C = out + 6*N;
    float* out_F = out + 15*N;
    #pragma unroll
    for (int d = 0; d < 3; ++d) {
        out_x[(size_t)p*3 + d] = xp[d] + DT_C * vnext[d];
        out_v[(size_t)p*3 + d] = vnext[d];
    }
    #pragma unroll
    for (int i = 0; i < 3; ++i)
        #pragma unroll
        for (int j = 0; j < 3; ++j) {
            out_C[(size_t)p*9 + i*3 + j] = Cn[i][j];
            out_F[(size_t)p*9 + i*3 + j] = Fn[i][j];
        }
}

extern "C" void kernel_launch(void* const* d_in, const int* in_sizes, int n_in,
                              void* d_out, int out_size, void* d_ws, size_t ws_size,
                              hipStream_t stream) {
    const float* x      = (const float*)d_in[0];
    const float* v      = (const float*)d_in[1];
    const float* C      = (const float*)d_in[2];
    const float* F      = (const float*)d_in[3];
    const float* stress = (const float*)d_in[4];
    float* out  = (float*)d_out;
    float* grid = (float*)d_ws;      // 128^3 * 4 f32 = 32 MB (fits in 192 MB L2)

    int n = in_sizes[0] / 3;

    const int blk = 256;             // 8 wave32 waves per block
    mpm_zero<<<NG3 / blk, blk, 0, stream>>>((float4*)grid);
    mpm_p2g<<<(n + blk - 1) / blk, blk, 0, stream>>>(x, v, C, stress, grid, n);
    mpm_grid_update<<<NG3 / blk, blk, 0, stream>>>((float4*)grid);
    mpm_g2p<<<(n + blk - 1) / blk, blk, 0, stream>>>(x, F, (const float4*)grid, out, n);
}